// NeuralODE_3478923510417
// MI455X (gfx1250) — compile-verified
//
#include <hip/hip_runtime.h>
#include <hip/hip_bf16.h>

typedef __attribute__((ext_vector_type(2))) float v2f;
typedef __attribute__((ext_vector_type(8))) float v8f;

#define TRAJ_LEN 25
#define N_SUB 8
#define NCOLS (2 * TRAJ_LEN)   // 50

// ---------------------------------------------------------------------------
// Kernel A: build the 25 propagator matrices P_k = (S^8)^k as a padded
// W[64][2] table in workspace (cols >= 50 are zero for the WMMA N-padding).
// The whole problem is linear: one fixed-step Tsit5 substep is y <- S*y with
// constant 2x2 S, so save point k is (S^8)^k applied to x0.
// ---------------------------------------------------------------------------
__device__ __forceinline__ void applyA(float a00, float a01, float a10, float a11,
                                       float y0, float y1, float& o0, float& o1) {
    o0 = fmaf(a00, y0, a01 * y1);
    o1 = fmaf(a10, y0, a11 * y1);
}

__global__ void build_props_kernel(const float* __restrict__ mat,
                                   const float* __restrict__ Tptr,
                                   float* __restrict__ W,
                                   float* __restrict__ num_steps_slot,
                                   int batch) {
    if (threadIdx.x != 0 || blockIdx.x != 0) return;

    const float T = Tptr[0];
    const float a00 = T * mat[0], a01 = T * mat[1];
    const float a10 = T * mat[2], a11 = T * mat[3];
    const float h = 1.0f / (float)((TRAJ_LEN - 1) * N_SUB);

    // Tsit5 tableau (float32, same as reference)
    const float A21 = 0.161f;
    const float A31 = -0.008480655492356989f, A32 = 0.335480655492357f;
    const float A41 = 2.8971530571054935f, A42 = -6.359448489975075f, A43 = 4.3622954328695815f;
    const float A51 = 5.325864828439257f, A52 = -11.748883564062828f,
                A53 = 7.4955393428898365f, A54 = -0.09249506636175525f;
    const float A61 = 5.86145544294642f, A62 = -12.92096931784711f,
                A63 = 8.159367898576159f, A64 = -0.071584973281401f,
                A65 = -0.028269050394068383f;
    const float B1 = 0.09646076681806523f, B2 = 0.01f, B3 = 0.4798896504144996f,
                B4 = 1.379008574103742f, B5 = -3.290069515436081f, B6 = 2.324710524099774f;

    // One substep's matrix S, column-by-column via basis vectors.
    float S[2][2];
    for (int c = 0; c < 2; ++c) {
        float y0 = (c == 0) ? 1.0f : 0.0f;
        float y1 = (c == 1) ? 1.0f : 0.0f;
        float k10, k11, k20, k21, k30, k31, k40, k41, k50, k51, k60, k61, t0, t1;
        applyA(a00, a01, a10, a11, y0, y1, k10, k11);
        t0 = y0 + h * (A21 * k10);
        t1 = y1 + h * (A21 * k11);
        applyA(a00, a01, a10, a11, t0, t1, k20, k21);
        t0 = y0 + h * (A31 * k10 + A32 * k20);
        t1 = y1 + h * (A31 * k11 + A32 * k21);
        applyA(a00, a01, a10, a11, t0, t1, k30, k31);
        t0 = y0 + h * (A41 * k10 + A42 * k20 + A43 * k30);
        t1 = y1 + h * (A41 * k11 + A42 * k21 + A43 * k31);
        applyA(a00, a01, a10, a11, t0, t1, k40, k41);
        t0 = y0 + h * (A51 * k10 + A52 * k20 + A53 * k30 + A54 * k40);
        t1 = y1 + h * (A51 * k11 + A52 * k21 + A53 * k31 + A54 * k41);
        applyA(a00, a01, a10, a11, t0, t1, k50, k51);
        t0 = y0 + h * (A61 * k10 + A62 * k20 + A63 * k30 + A64 * k40 + A65 * k50);
        t1 = y1 + h * (A61 * k11 + A62 * k21 + A63 * k31 + A64 * k41 + A65 * k51);
        applyA(a00, a01, a10, a11, t0, t1, k60, k61);
        S[0][c] = y0 + h * (B1 * k10 + B2 * k20 + B3 * k30 + B4 * k40 + B5 * k50 + B6 * k60);
        S[1][c] = y1 + h * (B1 * k11 + B2 * k21 + B3 * k31 + B4 * k41 + B5 * k51 + B6 * k61);
    }

    // M8 = S^8 (one save interval)
    float M8[2][2] = { {S[0][0], S[0][1]}, {S[1][0], S[1][1]} };
    for (int s = 1; s < N_SUB; ++s) {
        float n00 = S[0][0] * M8[0][0] + S[0][1] * M8[1][0];
        float n01 = S[0][0] * M8[0][1] + S[0][1] * M8[1][1];
        float n10 = S[1][0] * M8[0][0] + S[1][1] * M8[1][0];
        float n11 = S[1][0] * M8[0][1] + S[1][1] * M8[1][1];
        M8[0][0] = n00; M8[0][1] = n01; M8[1][0] = n10; M8[1][1] = n11;
    }

    // Column c = 2k + i of Out:  Out[b][c] = sum_j P_k[i][j] * x0[b][j].
    float P[2][2] = { {1.0f, 0.0f}, {0.0f, 1.0f} };
    for (int k = 0; k < TRAJ_LEN; ++k) {
        W[(2 * k + 0) * 2 + 0] = P[0][0];
        W[(2 * k + 0) * 2 + 1] = P[0][1];
        W[(2 * k + 1) * 2 + 0] = P[1][0];
        W[(2 * k + 1) * 2 + 1] = P[1][1];
        float n00 = M8[0][0] * P[0][0] + M8[0][1] * P[1][0];
        float n01 = M8[0][0] * P[0][1] + M8[0][1] * P[1][1];
        float n10 = M8[1][0] * P[0][0] + M8[1][1] * P[1][0];
        float n11 = M8[1][0] * P[0][1] + M8[1][1] * P[1][1];
        P[0][0] = n00; P[0][1] = n01; P[1][0] = n10; P[1][1] = n11;
    }
    // zero the N-padding columns (50..63); rewritten deterministically per call
    for (int c = NCOLS; c < 64; ++c) {
        W[c * 2 + 0] = 0.0f;
        W[c * 2 + 1] = 0.0f;
    }

    // num_steps = batch * 24 * 8, bit-cast int32 into the trailing output slot
    *((int*)num_steps_slot) = batch * (TRAJ_LEN - 1) * N_SUB;
}

// ---------------------------------------------------------------------------
// Kernel B: Out[B,50] = X0[B,2(pad 4)] @ W^T via V_WMMA_F32_16X16X4_F32.
// One wave handles exactly 16 batch rows (host guarantees full tiles).
// No divergent row guards: straight-line loads -> 4 WMMAs -> store stream.
// ---------------------------------------------------------------------------
__global__ void ode_wmma_kernel(const float* __restrict__ x0,
                                const float* __restrict__ W,
                                float* __restrict__ out,
                                int batch16) {
    const int lane = (int)(threadIdx.x & 31);
    const int wave = (int)((blockIdx.x * blockDim.x + threadIdx.x) >> 5);
    const int rowBase = wave * 16;
    if (rowBase >= batch16) return;   // uniform per-wave

    const int  l15 = lane & 15;
    const bool lo  = lane < 16;

    // A fragment (16x4 f32): lanes 0-15 carry x0 (K=0,1); lanes 16-31 carry
    // the K=2,3 zero padding. Unconditional load + cndmask select (no exec
    // manipulation); lanes 16-31 redundantly load the same safe addresses.
    const float2 av = ((const float2*)x0)[rowBase + l15];
    v2f a;
    a.x = lo ? av.x : 0.0f;
    a.y = lo ? av.y : 0.0f;

#pragma unroll
    for (int nt = 0; nt < 4; ++nt) {
        const int col = nt * 16 + l15;

        // B fragment (4x16 f32): lanes 0-15 = W[col][0..1] (K=0,1);
        // lanes 16-31 = zero padding (K=2,3). col <= 63 always in-bounds
        // of the padded 64-entry table.
        const float2 wv = ((const float2*)W)[col];
        v2f b;
        b.x = lo ? wv.x : 0.0f;
        b.y = lo ? wv.y : 0.0f;

        v8f d = {};
        // (neg_a, A, neg_b, B, c_mod, C, reuse_a, reuse_b)
        d = __builtin_amdgcn_wmma_f32_16x16x4_f32(
            false, a, false, b, (short)0, d, false, false);

        // D layout: VGPR r -> row rOff+r; column = lane%16 within this N-tile.
        // One base pointer per tile; consecutive rows become immediate offsets.
        if (col < NCOLS) {
            const int rOff = (lane >> 4) * 8;
            float* p = out + (size_t)(rowBase + rOff) * NCOLS + col;
#pragma unroll
            for (int r = 0; r < 8; ++r) {
                p[r * NCOLS] = d[r];
            }
        }
    }
}

// Scalar tail for batch % 16 rows (not launched for the reference shape).
__global__ void ode_tail_kernel(const float* __restrict__ x0,
                                const float* __restrict__ W,
                                float* __restrict__ out,
                                int start, int batch) {
    const int row = start + (int)(blockIdx.x * blockDim.x + threadIdx.x);
    if (row >= batch) return;
    const float y0 = x0[row * 2 + 0];
    const float y1 = x0[row * 2 + 1];
    for (int c = 0; c < NCOLS; ++c) {
        out[(size_t)row * NCOLS + c] = fmaf(W[c * 2 + 0], y0, W[c * 2 + 1] * y1);
    }
}

extern "C" void kernel_launch(void* const* d_in, const int* in_sizes, int n_in,
                              void* d_out, int out_size, void* d_ws, size_t ws_size,
                              hipStream_t stream) {
    const float* x0  = (const float*)d_in[0];   // [B,2]
    const float* mat = (const float*)d_in[1];   // [2,2]
    const float* Tp  = (const float*)d_in[2];   // scalar
    float* out = (float*)d_out;                 // [B,25,2] floats + 1 int slot
    float* W   = (float*)d_ws;                  // padded 64 x 2 f32 table

    const int batch   = in_sizes[0] / 2;
    const int batch16 = batch & ~15;            // full 16-row WMMA tiles

    build_props_kernel<<<1, 32, 0, stream>>>(mat, Tp, W, out + (out_size - 1), batch);

    if (batch16 > 0) {
        const int waves   = batch16 / 16;
        const int threads = 256;                 // 8 waves per block
        const int blocks  = (waves * 32 + threads - 1) / threads;
        ode_wmma_kernel<<<blocks, threads, 0, stream>>>(x0, W, out, batch16);
    }
    if (batch16 < batch) {
        const int rem = batch - batch16;
        ode_tail_kernel<<<(rem + 255) / 256, 256, 0, stream>>>(x0, W, out, batch16, batch);
    }
}